// Generator_21947282882695
// MI455X (gfx1250) — compile-verified
//
#include <hip/hip_runtime.h>
#include <math.h>

#define HID     1024
#define HID3    3072
#define SRC_LEN 256
#define MAX_LEN 128
#define OUT_DIM 32000
#define SOS     1
#define EOS     2
#define NB_LOG  1000   // logits kernel blocks: 32 rows per block -> 32000 rows

typedef float v2f __attribute__((ext_vector_type(2)));
typedef float v8f __attribute__((ext_vector_type(8)));

struct DecState { float M, S; int mask, tok, done; };

// ---------------- wave32 helpers ----------------
__device__ __forceinline__ float wred(float v) {
#pragma unroll
    for (int off = 16; off; off >>= 1) v += __shfl_xor(v, off, 32);
    return v;   // all lanes hold the total (butterfly)
}

__device__ __forceinline__ float sigmoidf_(float x) {
    return 1.0f / (1.0f + expf(-x));
}

// ---------------------------------------------------------------------------
// Kernel 1: encoder input-gate GEMM with f32 WMMA.
//   GI[t][m] = sum_k enc_w_ih[m][k] * enc_emb[X[t]][k] + b_ih[m]
// One wave (32 lanes) per 16x16 output tile; K marched in steps of 4 via
// V_WMMA_F32_16X16X4_F32. A layout: lane=M (0..15), half-wave selects K pair.
// D layout: VGPR v -> (M = v + 8*(lane>>4), N = lane&15).
// ---------------------------------------------------------------------------
__global__ void enc_gi_gemm(const int* __restrict__ X,
                            const float* __restrict__ emb,
                            const float* __restrict__ w_ih,
                            const float* __restrict__ b_ih,
                            float* __restrict__ GI)
{
    const int lane = threadIdx.x;            // 0..31 (blockDim.x == 32)
    const int m0   = blockIdx.x * 16;        // row tile within [0,3072)
    const int n0   = blockIdx.y * 16;        // token tile within [0,256)
    const int mloc = lane & 15;
    const int half = lane >> 4;              // 0 -> K{0,1}, 1 -> K{2,3}
    const int col  = n0 + mloc;              // this lane's B column (token)
    const int tok  = X[col];

    const float* arow = w_ih + (size_t)(m0 + mloc) * HID + 2 * half;
    const float* bcol = emb  + (size_t)tok * HID + 2 * half;

    v8f c;
#pragma unroll
    for (int v = 0; v < 8; ++v) c[v] = b_ih[m0 + v + 8 * half];

#pragma unroll 4
    for (int k = 0; k < HID; k += 4) {
        v2f a = *reinterpret_cast<const v2f*>(arow + k);
        v2f b = *reinterpret_cast<const v2f*>(bcol + k);
        c = __builtin_amdgcn_wmma_f32_16x16x4_f32(
                /*neg_a=*/false, a, /*neg_b=*/false, b,
                /*c_mod=*/(short)0, c, /*reuse_a=*/false, /*reuse_b=*/false);
    }

#pragma unroll
    for (int v = 0; v < 8; ++v)
        GI[(size_t)col * HID3 + (m0 + v + 8 * half)] = c[v];
}

// ---------------------------------------------------------------------------
// Kernel 2: one encoder GRU recurrence step (1024 waves, one output elem each).
//   gh = w_hh @ h + b_hh ; gate math with precomputed gi row.
// ---------------------------------------------------------------------------
__global__ void gru_enc_step(const float* __restrict__ gi,     // GI + t*3072
                             const float* __restrict__ w_hh,
                             const float* __restrict__ b_hh,
                             const float* __restrict__ h_in,
                             float* __restrict__ h_out)
{
    const int wave = (blockIdx.x * blockDim.x + threadIdx.x) >> 5;  // 0..1023
    const int lane = threadIdx.x & 31;

    const float4* h4 = reinterpret_cast<const float4*>(h_in);
    float4 hl[8];
#pragma unroll
    for (int i = 0; i < 8; ++i) hl[i] = h4[lane + i * 32];

    float gh[3];
#pragma unroll
    for (int g = 0; g < 3; ++g) {
        const int row = wave + g * HID;
        const float4* w4 = reinterpret_cast<const float4*>(w_hh + (size_t)row * HID);
        float acc = 0.0f;
#pragma unroll
        for (int i = 0; i < 8; ++i) {
            float4 w = w4[lane + i * 32];
            acc += w.x * hl[i].x + w.y * hl[i].y + w.z * hl[i].z + w.w * hl[i].w;
        }
        acc = wred(acc);
        gh[g] = acc + b_hh[row];
    }

    if (lane == 0) {
        const int j = wave;
        float r = sigmoidf_(gi[j]           + gh[0]);
        float z = sigmoidf_(gi[j + HID]     + gh[1]);
        float n = tanhf    (gi[j + 2 * HID] + r * gh[2]);
        h_out[j] = (1.0f - z) * n + z * h_in[j];
    }
}

// ---------------------------------------------------------------------------
// Kernel 3: one decoder GRU step. x = relu(dec_emb[tok]); two GEMVs + gates.
// Writes h_new (commit is masked later by `done`).
// ---------------------------------------------------------------------------
__global__ void gru_dec_step(const DecState* __restrict__ st,
                             const float* __restrict__ dec_emb,
                             const float* __restrict__ w_ih,
                             const float* __restrict__ b_ih,
                             const float* __restrict__ w_hh,
                             const float* __restrict__ b_hh,
                             const float* __restrict__ h_in,
                             float* __restrict__ h_new)
{
    const int wave = (blockIdx.x * blockDim.x + threadIdx.x) >> 5;  // 0..1023
    const int lane = threadIdx.x & 31;
    const int tok  = st->tok;

    const float4* x4 = reinterpret_cast<const float4*>(dec_emb + (size_t)tok * HID);
    const float4* h4 = reinterpret_cast<const float4*>(h_in);
    float4 xl[8], hl[8];
#pragma unroll
    for (int i = 0; i < 8; ++i) {
        float4 x = x4[lane + i * 32];
        xl[i] = make_float4(fmaxf(x.x, 0.f), fmaxf(x.y, 0.f),
                            fmaxf(x.z, 0.f), fmaxf(x.w, 0.f));
        hl[i] = h4[lane + i * 32];
    }

    float gi[3], gh[3];
#pragma unroll
    for (int g = 0; g < 3; ++g) {
        const int row = wave + g * HID;
        const float4* wi4 = reinterpret_cast<const float4*>(w_ih + (size_t)row * HID);
        const float4* wh4 = reinterpret_cast<const float4*>(w_hh + (size_t)row * HID);
        float ai = 0.0f, ah = 0.0f;
#pragma unroll
        for (int i = 0; i < 8; ++i) {
            float4 wi = wi4[lane + i * 32];
            float4 wh = wh4[lane + i * 32];
            ai += wi.x * xl[i].x + wi.y * xl[i].y + wi.z * xl[i].z + wi.w * xl[i].w;
            ah += wh.x * hl[i].x + wh.y * hl[i].y + wh.z * hl[i].z + wh.w * hl[i].w;
        }
        gi[g] = wred(ai) + b_ih[row];
        gh[g] = wred(ah) + b_hh[row];
    }

    if (lane == 0) {
        const int j = wave;
        float r = sigmoidf_(gi[0] + gh[0]);
        float z = sigmoidf_(gi[1] + gh[1]);
        float n = tanhf    (gi[2] + r * gh[2]);
        h_new[j] = (1.0f - z) * n + z * h_in[j];
    }
}

// ---------------------------------------------------------------------------
// Kernel 4: logits GEMV (32000x1024) fused with online (max, sumexp, argmax).
// 1000 blocks x 8 waves x 4 rows. lin_w (131 MB) is L2-resident on MI455X.
// ---------------------------------------------------------------------------
__global__ void logits_kernel(const float* __restrict__ lin_w,
                              const float* __restrict__ lin_b,
                              const float* __restrict__ h,
                              float* __restrict__ logits,
                              float* __restrict__ pmax,
                              float* __restrict__ psum,
                              int*   __restrict__ pidx)
{
    __shared__ float sm[8], ss[8];
    __shared__ int   si[8];

    const int lane = threadIdx.x & 31;
    const int wv   = threadIdx.x >> 5;

    const float4* h4 = reinterpret_cast<const float4*>(h);
    float4 hl[8];
#pragma unroll
    for (int i = 0; i < 8; ++i) hl[i] = h4[lane + i * 32];

    float m = -INFINITY, s = 0.0f;
    int   idx = 0;
    const int base = (blockIdx.x * 8 + wv) * 4;

#pragma unroll
    for (int r = 0; r < 4; ++r) {
        const int row = base + r;
        const float4* w4 = reinterpret_cast<const float4*>(lin_w + (size_t)row * HID);
        float acc = 0.0f;
#pragma unroll
        for (int i = 0; i < 8; ++i) {
            float4 w = w4[lane + i * 32];
            acc += w.x * hl[i].x + w.y * hl[i].y + w.z * hl[i].z + w.w * hl[i].w;
        }
        acc = wred(acc);                       // all lanes get the total
        const float logit = acc + lin_b[row];
        if (lane == 0) logits[row] = logit;
        if (logit > m) { s = s * expf(m - logit) + 1.0f; m = logit; idx = row; }
        else           { s += expf(logit - m); }
    }

    if (lane == 0) { sm[wv] = m; ss[wv] = s; si[wv] = idx; }
    __syncthreads();

    if (threadIdx.x == 0) {
        float M = sm[0], S = ss[0]; int I = si[0];
        for (int w = 1; w < 8; ++w) {
            const float m2 = sm[w], s2 = ss[w]; const int i2 = si[w];
            const float Mn = fmaxf(M, m2);
            S = S * expf(M - Mn) + s2 * expf(m2 - Mn);
            if (m2 > M || (m2 == M && i2 < I)) I = i2;
            M = Mn;
        }
        pmax[blockIdx.x] = M; psum[blockIdx.x] = S; pidx[blockIdx.x] = I;
    }
}

// ---------------------------------------------------------------------------
// Kernel 5: reduce block partials -> global (max, sumexp, argmax); greedy-step
// state machine (done / token carry) + write the output token.
// ---------------------------------------------------------------------------
__global__ void argmax_reduce(const float* __restrict__ pmax,
                              const float* __restrict__ psum,
                              const int*   __restrict__ pidx,
                              DecState* __restrict__ st,
                              float* __restrict__ out_seq,
                              int step)
{
    __shared__ float sm[256], ss[256];
    __shared__ int   si[256];
    const int t = threadIdx.x;

    float M = -INFINITY, S = 0.0f; int I = 0x7fffffff;
    for (int p = t; p < NB_LOG; p += 256) {
        const float m2 = pmax[p], s2 = psum[p]; const int i2 = pidx[p];
        const float Mn = fmaxf(M, m2);
        S = S * expf(M - Mn) + s2 * expf(m2 - Mn);
        if (m2 > M || (m2 == M && i2 < I)) I = i2;
        M = Mn;
    }
    sm[t] = M; ss[t] = S; si[t] = I;
    __syncthreads();

    for (int stride = 128; stride > 0; stride >>= 1) {
        if (t < stride) {
            const float M1 = sm[t], S1 = ss[t]; const int I1 = si[t];
            const float m2 = sm[t + stride], s2 = ss[t + stride];
            const int   i2 = si[t + stride];
            const float Mn = fmaxf(M1, m2);
            ss[t] = S1 * expf(M1 - Mn) + s2 * expf(m2 - Mn);
            sm[t] = Mn;
            si[t] = (m2 > M1 || (m2 == M1 && i2 < I1)) ? i2 : I1;
        }
        __syncthreads();
    }

    if (t == 0) {
        const int nxt     = si[0];
        const int done_in = st->done;
        const int tok_in  = st->tok;
        const int out_tok = done_in ? EOS : nxt;
        out_seq[step + 1] = (float)out_tok;
        st->mask = done_in;                       // mask for probs of THIS step
        st->tok  = done_in ? tok_in : nxt;
        st->done = done_in | (nxt == EOS);
        st->M = sm[0]; st->S = ss[0];
    }
}

// ---------------------------------------------------------------------------
// Kernel 6: write probs row (zeroed if carried-done) and commit h <- h_new.
// ---------------------------------------------------------------------------
__global__ void probs_commit(const float* __restrict__ logits,
                             const DecState* __restrict__ st,
                             const float* __restrict__ h_new,
                             float* __restrict__ h,
                             float* __restrict__ out_probs)
{
    const int gid = blockIdx.x * blockDim.x + threadIdx.x;
    const float M = st->M, S = st->S;
    const int mask = st->mask;
    if (gid < OUT_DIM)
        out_probs[gid] = mask ? 0.0f : expf(logits[gid] - M) / S;
    if (gid < HID && !mask)
        h[gid] = h_new[gid];
}

// ---------------------------------------------------------------------------
// Kernel 0: init hidden state, decode state, SOS output token.
// ---------------------------------------------------------------------------
__global__ void init_kernel(float* __restrict__ hA,
                            DecState* __restrict__ st,
                            float* __restrict__ out_seq)
{
    const int gid = blockIdx.x * blockDim.x + threadIdx.x;
    if (gid < HID) hA[gid] = 0.0f;
    if (gid == 0) {
        st->tok = SOS; st->done = 0; st->mask = 0; st->M = 0.f; st->S = 1.f;
        out_seq[0] = (float)SOS;
    }
}

// ---------------------------------------------------------------------------
extern "C" void kernel_launch(void* const* d_in, const int* in_sizes, int n_in,
                              void* d_out, int out_size, void* d_ws, size_t ws_size,
                              hipStream_t stream)
{
    (void)in_sizes; (void)n_in; (void)out_size; (void)ws_size;

    const int*   X        = (const int*)  d_in[0];
    const float* enc_emb  = (const float*)d_in[1];
    const float* enc_w_ih = (const float*)d_in[2];
    const float* enc_w_hh = (const float*)d_in[3];
    const float* enc_b_ih = (const float*)d_in[4];
    const float* enc_b_hh = (const float*)d_in[5];
    const float* dec_emb  = (const float*)d_in[6];
    const float* dec_w_ih = (const float*)d_in[7];
    const float* dec_w_hh = (const float*)d_in[8];
    const float* dec_b_ih = (const float*)d_in[9];
    const float* dec_b_hh = (const float*)d_in[10];
    const float* lin_w    = (const float*)d_in[11];
    const float* lin_b    = (const float*)d_in[12];

    float* out = (float*)d_out;    // [0..128] token seq, then 128*32000 probs

    float* ws = (float*)d_ws;
    float* GI     = ws;  ws += (size_t)SRC_LEN * HID3;   // 786432
    float* hA     = ws;  ws += HID;
    float* hB     = ws;  ws += HID;
    float* hN     = ws;  ws += HID;
    float* logits = ws;  ws += OUT_DIM;
    float* pmax   = ws;  ws += NB_LOG;
    float* psum   = ws;  ws += NB_LOG;
    int*   pidx   = (int*)ws;  ws += NB_LOG;
    DecState* st  = (DecState*)ws;

    init_kernel<<<4, 256, 0, stream>>>(hA, st, out);

    // Encoder input gates for all 256 tokens in one WMMA GEMM.
    enc_gi_gemm<<<dim3(HID3 / 16, SRC_LEN / 16), 32, 0, stream>>>(
        X, enc_emb, enc_w_ih, enc_b_ih, GI);

    // Encoder recurrence (ping-pong h between hA/hB; SRC_LEN even -> ends in hA).
    for (int t = 0; t < SRC_LEN; ++t) {
        const float* hin = (t & 1) ? hB : hA;
        float*       hout= (t & 1) ? hA : hB;
        gru_enc_step<<<128, 256, 0, stream>>>(GI + (size_t)t * HID3,
                                              enc_w_hh, enc_b_hh, hin, hout);
    }

    // Greedy decoder: hA carries committed hidden state.
    for (int s = 0; s < MAX_LEN; ++s) {
        gru_dec_step<<<128, 256, 0, stream>>>(st, dec_emb, dec_w_ih, dec_b_ih,
                                              dec_w_hh, dec_b_hh, hA, hN);
        logits_kernel<<<NB_LOG, 256, 0, stream>>>(lin_w, lin_b, hN,
                                                  logits, pmax, psum, pidx);
        argmax_reduce<<<1, 256, 0, stream>>>(pmax, psum, pidx, st, out, s);
        probs_commit<<<125, 256, 0, stream>>>(logits, st, hN, hA,
                                              out + 129 + (size_t)s * OUT_DIM);
    }
}